// Repair_21543555957426
// MI455X (gfx1250) — compile-verified
//
#include <hip/hip_runtime.h>
#include <hip/hip_bf16.h>
#include <math.h>

#define D 64
#define NNODES 10000
#define NEDGES 160000
#define RREM 16
#define CC 512

typedef __attribute__((ext_vector_type(2))) float v2f;
typedef __attribute__((ext_vector_type(8))) float v8f;

// ---------------------------------------------------------------------------
// h[n,d] = coords[n,0]*W_emb[0,d] + coords[n,1]*W_emb[1,d] + b_emb[d]
// ---------------------------------------------------------------------------
__global__ void embed_kernel(const float* __restrict__ coords,
                             const float* __restrict__ W_emb,
                             const float* __restrict__ b_emb,
                             float* __restrict__ h) {
  int idx = blockIdx.x * blockDim.x + threadIdx.x;
  if (idx >= NNODES * D) return;
  int n = idx >> 6, d = idx & 63;
  float c0 = coords[2 * n], c1 = coords[2 * n + 1];
  h[idx] = fmaf(c0, W_emb[d], fmaf(c1, W_emb[D + d], b_emb[d]));
}

__global__ void zero_kernel(float* __restrict__ p, int n) {
  int i = blockIdx.x * blockDim.x + threadIdx.x;
  if (i < n) p[i] = 0.0f;
}

// ---------------------------------------------------------------------------
// Masked neighbor aggregation: agg[dst] += h[src] for unmasked edges,
// deg[dst] += 1. Thread = (edge, dim) -> coalesced loads & atomics.
// ---------------------------------------------------------------------------
__global__ void scatter_kernel(const float* __restrict__ h,
                               const int* __restrict__ edge_src,
                               const int* __restrict__ edge_dst,
                               const int* __restrict__ removal_idx, int r,
                               float* __restrict__ agg,
                               float* __restrict__ deg) {
  int idx = blockIdx.x * blockDim.x + threadIdx.x;
  int e = idx >> 6, d = idx & 63;
  if (e >= NEDGES) return;
  int rm = removal_idx[r];
  int s = edge_src[e], t = edge_dst[e];
  if (s == rm || t == rm) return;  // emask == 0
  atomicAdd(&agg[t * D + d], h[s * D + d]);
  if (d == 0) atomicAdd(&deg[t], 1.0f);
}

// ---------------------------------------------------------------------------
// h = relu(h@Ws + (agg/max(deg,1))@Wn + b) + h     (in place, 16-row tiles)
// 4 waves / block; wave w owns 16-column tile w. fp32 WMMA 16x16x4, K=64.
// ---------------------------------------------------------------------------
__global__ void __launch_bounds__(128)
gnn_gemm_kernel(float* __restrict__ h, const float* __restrict__ agg,
                const float* __restrict__ deg,
                const float* __restrict__ Ws, const float* __restrict__ Wn,
                const float* __restrict__ bvec) {
  __shared__ float sH[16][D + 1];  // +1 pad: kill 16-way bank conflicts
  __shared__ float sG[16][D + 1];
  __shared__ float sDeg[16];

  int rowbase = blockIdx.x * 16;
  for (int i = threadIdx.x; i < 16 * D; i += 128) {
    int rr = i >> 6, cc = i & 63;
    sH[rr][cc] = h[(rowbase + rr) * D + cc];
    sG[rr][cc] = agg[(rowbase + rr) * D + cc];
  }
  if (threadIdx.x < 16) sDeg[threadIdx.x] = deg[rowbase + threadIdx.x];
  __syncthreads();

  int lane = threadIdx.x & 31;
  int wave = threadIdx.x >> 5;   // column tile 0..3
  int col0 = wave * 16;
  int M    = lane & 15;          // A row / B,D column-in-tile
  int half = lane >> 4;
  int colg = col0 + M;           // this lane's global output column

  float inv = 1.0f / fmaxf(sDeg[M], 1.0f);  // 1/deg for this lane's A row

  v8f acc = {0.f, 0.f, 0.f, 0.f, 0.f, 0.f, 0.f, 0.f};
#pragma unroll
  for (int kk = 0; kk < 16; ++kk) {
    int k0 = kk * 4 + 2 * half;  // A lane holds K = k0, k0+1 (ISA 16x4 layout)
    v2f a, b;
    // --- h @ Ws ---
    a.x = sH[M][k0];
    a.y = sH[M][k0 + 1];
    b.x = Ws[k0 * D + colg];
    b.y = Ws[(k0 + 1) * D + colg];
    acc = __builtin_amdgcn_wmma_f32_16x16x4_f32(false, a, false, b,
                                                (short)0, acc, false, false);
    // --- (agg/deg) @ Wn ---
    a.x = sG[M][k0] * inv;
    a.y = sG[M][k0 + 1] * inv;
    b.x = Wn[k0 * D + colg];
    b.y = Wn[(k0 + 1) * D + colg];
    acc = __builtin_amdgcn_wmma_f32_16x16x4_f32(false, a, false, b,
                                                (short)0, acc, false, false);
  }

  float bias = bvec[colg];
#pragma unroll
  for (int j = 0; j < 8; ++j) {
    int mo = j + 8 * half;  // C/D layout: VGPR j, lane half -> output row
    float v = fmaxf(acc[j] + bias, 0.0f) + sH[mo][colg];  // relu + residual
    h[(rowbase + mo) * D + colg] = v;
  }
}

// ---------------------------------------------------------------------------
// Per scored edge c: ef = relu([h[s]||h[t]] @ W_edge + b_edge)
// score[c] = [r_embed || ef] @ W_score + b_score
// ---------------------------------------------------------------------------
__global__ void __launch_bounds__(64)
edge_score_kernel(const float* __restrict__ h, const float* __restrict__ coords,
                  const float* __restrict__ W_emb, const float* __restrict__ b_emb,
                  const float* __restrict__ W_edge, const float* __restrict__ b_edge,
                  const float* __restrict__ W_score, const float* __restrict__ b_score,
                  const int* __restrict__ edge_src, const int* __restrict__ edge_dst,
                  const int* __restrict__ removal_idx, const int* __restrict__ conn_eidx,
                  int r, float* __restrict__ score) {
  int c = blockIdx.x;
  int d = threadIdx.x;  // 0..63
  int rm = removal_idx[r];
  int e = conn_eidx[r * CC + c];
  int s = edge_src[e], t = edge_dst[e];
  const float* hs = h + s * D;
  const float* ht = h + t * D;

  float acc = b_edge[d];
#pragma unroll 4
  for (int j = 0; j < D; ++j) {
    acc = fmaf(hs[j], W_edge[j * D + d], acc);
    acc = fmaf(ht[j], W_edge[(D + j) * D + d], acc);
  }
  float ef = fmaxf(acc, 0.0f);
  float remb = fmaf(coords[2 * rm], W_emb[d],
                    fmaf(coords[2 * rm + 1], W_emb[D + d], b_emb[d]));
  float part = ef * W_score[D + d] + remb * W_score[d];

  __shared__ float red[64];
  red[d] = part;
  __syncthreads();
  for (int off = 32; off > 0; off >>= 1) {
    if (d < off) red[d] += red[d + off];
    __syncthreads();
  }
  if (d == 0) score[c] = red[0] + b_score[0];
}

// ---------------------------------------------------------------------------
// softmax over C=512 scores -> out[c]
// ---------------------------------------------------------------------------
__global__ void __launch_bounds__(CC)
softmax_kernel(const float* __restrict__ score, float* __restrict__ out) {
  __shared__ float red[CC];
  int i = threadIdx.x;
  float v = score[i];
  red[i] = v;
  __syncthreads();
  for (int off = CC / 2; off > 0; off >>= 1) {
    if (i < off) red[i] = fmaxf(red[i], red[i + off]);
    __syncthreads();
  }
  float mx = red[0];
  __syncthreads();
  float e = expf(v - mx);
  red[i] = e;
  __syncthreads();
  for (int off = CC / 2; off > 0; off >>= 1) {
    if (i < off) red[i] += red[i + off];
    __syncthreads();
  }
  out[i] = e / red[0];
}

// ---------------------------------------------------------------------------
extern "C" void kernel_launch(void* const* d_in, const int* in_sizes, int n_in,
                              void* d_out, int out_size, void* d_ws, size_t ws_size,
                              hipStream_t stream) {
  const float* coords   = (const float*)d_in[0];
  const float* W_emb    = (const float*)d_in[1];
  const float* b_emb    = (const float*)d_in[2];
  const float* W_self0  = (const float*)d_in[3];
  const float* W_neigh0 = (const float*)d_in[4];
  const float* b_gnn0   = (const float*)d_in[5];
  const float* W_self1  = (const float*)d_in[6];
  const float* W_neigh1 = (const float*)d_in[7];
  const float* b_gnn1   = (const float*)d_in[8];
  const float* W_edge   = (const float*)d_in[9];
  const float* b_edge   = (const float*)d_in[10];
  const float* W_score  = (const float*)d_in[11];
  const float* b_score  = (const float*)d_in[12];
  const int*   edge_src = (const int*)d_in[13];
  const int*   edge_dst = (const int*)d_in[14];
  const int*   removal_idx = (const int*)d_in[15];
  const int*   conn_eidx   = (const int*)d_in[16];
  float* out = (float*)d_out;

  float* ws    = (float*)d_ws;
  float* h     = ws;                  // N*D
  float* agg   = h + NNODES * D;      // N*D   (deg must follow: zeroed together)
  float* deg   = agg + NNODES * D;    // N
  float* score = deg + NNODES;        // C

  for (int r = 0; r < RREM; ++r) {
    embed_kernel<<<(NNODES * D + 255) / 256, 256, 0, stream>>>(coords, W_emb,
                                                               b_emb, h);
    for (int layer = 0; layer < 2; ++layer) {
      const float* Ws = layer ? W_self1 : W_self0;
      const float* Wn = layer ? W_neigh1 : W_neigh0;
      const float* bg = layer ? b_gnn1 : b_gnn0;
      zero_kernel<<<(NNODES * D + NNODES + 255) / 256, 256, 0, stream>>>(
          agg, NNODES * D + NNODES);
      scatter_kernel<<<(NEDGES * 64) / 256, 256, 0, stream>>>(
          h, edge_src, edge_dst, removal_idx, r, agg, deg);
      gnn_gemm_kernel<<<NNODES / 16, 128, 0, stream>>>(h, agg, deg, Ws, Wn, bg);
    }
    edge_score_kernel<<<CC, 64, 0, stream>>>(
        h, coords, W_emb, b_emb, W_edge, b_edge, W_score, b_score,
        edge_src, edge_dst, removal_idx, conn_eidx, r, score);
    softmax_kernel<<<1, CC, 0, stream>>>(score, out + r * CC);
  }
}